// SepFormerArtifactNet_52810917871661
// MI455X (gfx1250) — compile-verified
//
#include <hip/hip_runtime.h>
#include <hip/hip_bf16.h>
#include <math.h>

// ---------------------------------------------------------------------------
// SepFormer forward on gfx1250 (CDNA5), fp32 WMMA (V_WMMA_F32_16X16X4_F32).
// Round 3: async global->LDS weight staging (ASYNCcnt path), branch-free
// software-pipelined A loads (peeled final k-step).
// ---------------------------------------------------------------------------

typedef float v2f __attribute__((ext_vector_type(2)));
typedef float v8f __attribute__((ext_vector_type(8)));

__device__ __forceinline__ v8f wmma4(v2f a, v2f b, v8f c) {
  // 8 args: (neg_a, A, neg_b, B, c_mod, C, reuse_a, reuse_b)
  return __builtin_amdgcn_wmma_f32_16x16x4_f32(false, a, false, b, (short)0, c,
                                               false, false);
}

// Model constants
#define SDIM 111
#define KDIM 64
#define POSN 7104          // 111*64
#define NTOK 28416         // 444*64 == 256*111
#define CDIM 128
#define LCONV 3583
#define QS 36              // LDS row stride for Q/K (bank-conflict-free frags)
#define VS 40              // LDS row stride for V

// ---------------------------------------------------------------------------
// 1) Encoder: conv1d(1->128, ks=16, stride=8, VALID) + ReLU, written directly
//    into chunked (b, c, s, k) layout; pad region (pos >= 3583) -> 0.
// ---------------------------------------------------------------------------
__global__ void encoder_kernel(const float* __restrict__ x,
                               const float* __restrict__ w,
                               float* __restrict__ state) {
  int p = blockIdx.x;              // (b*111 + s)*64 + k  in [0, 28416)
  int c = threadIdx.x;             // channel 0..127
  int b = p / POSN;
  int rem = p % POSN;
  int s = rem / KDIM;
  int k = rem % KDIM;
  int pos = s * 32 + k;            // HOP=32
  __shared__ float win[16];
  float val = 0.f;
  if (pos < LCONV) {
    if (c < 16) win[c] = x[(size_t)b * 28672 + (size_t)pos * 8 + c];
    __syncthreads();
    float acc = 0.f;
#pragma unroll
    for (int t = 0; t < 16; ++t) acc += w[c * 16 + t] * win[t];
    val = fmaxf(acc, 0.f);
  } else {
    __syncthreads();
  }
  state[((size_t)b * CDIM + c) * POSN + (size_t)s * KDIM + k] = val;
}

// ---------------------------------------------------------------------------
// 2) Gather (b,c,s,k) state -> token-major (Bt, T, C) + sinusoidal PE.
// ---------------------------------------------------------------------------
__global__ void gather_pe_kernel(const float* __restrict__ state,
                                 float* __restrict__ tok, int total, int intra) {
  int e = blockIdx.x * blockDim.x + threadIdx.x;
  if (e >= total) return;
  int token = e >> 7;
  int c = e & 127;
  int b = token / POSN;
  int rem = token % POSN;
  int s, k, pos;
  if (intra) { s = rem / KDIM; k = rem % KDIM; pos = k; }
  else       { k = rem / SDIM; s = rem % SDIM; pos = s; }
  float freq = expf((float)(c & ~1) * (-9.210340371976184f / 128.f)); // -ln(1e4)/d
  float ang = (float)pos * freq;
  float pe = (c & 1) ? cosf(ang) : sinf(ang);
  tok[e] = state[((size_t)b * CDIM + c) * POSN + (size_t)s * KDIM + k] + pe;
}

// ---------------------------------------------------------------------------
// 3) LayerNorm over last dim (128), one wave32 per token, float4 per lane.
// ---------------------------------------------------------------------------
__global__ void ln_kernel(const float* __restrict__ x, const float* __restrict__ g,
                          const float* __restrict__ b, float* __restrict__ y,
                          int ntok) {
  int wave = (blockIdx.x * blockDim.x + threadIdx.x) >> 5;
  int lane = threadIdx.x & 31;
  if (wave >= ntok) return;
  float4 v = ((const float4*)(x + (size_t)wave * CDIM))[lane];
  float s = v.x + v.y + v.z + v.w;
#pragma unroll
  for (int o = 16; o; o >>= 1) s += __shfl_xor(s, o, 32);
  float mean = s * (1.f / 128.f);
  float dx = v.x - mean, dy = v.y - mean, dz = v.z - mean, dw = v.w - mean;
  float q = dx * dx + dy * dy + dz * dz + dw * dw;
#pragma unroll
  for (int o = 16; o; o >>= 1) q += __shfl_xor(q, o, 32);
  float rstd = rsqrtf(q * (1.f / 128.f) + 1e-5f);
  float4 gv = ((const float4*)g)[lane];
  float4 bv = ((const float4*)b)[lane];
  float4 out;
  out.x = dx * rstd * gv.x + bv.x;
  out.y = dy * rstd * gv.y + bv.y;
  out.z = dz * rstd * gv.z + bv.z;
  out.w = dw * rstd * gv.w + bv.w;
  ((float4*)(y + (size_t)wave * CDIM))[lane] = out;
}

// ---------------------------------------------------------------------------
// 4) WMMA GEMM:  out(N x M) (=|+=) act(N x K) @ W(M x K)^T + bias
//    Block = 8 waves sharing one 64-feature weight slab, staged into LDS via
//    GLOBAL_LOAD_ASYNC_TO_LDS_B128 (ASYNCcnt) in 16B chunks that respect the
//    K+4 anti-bank-conflict row padding. Wave tile = 16 tokens x 64 features;
//    A loads pipelined one k-step ahead, last step peeled (branch-free loop).
//    Requires (ntok/16) % 8 == 0 (28416/16 = 1776 = 8*222).
// ---------------------------------------------------------------------------
__global__ void gemm_kernel(const float* __restrict__ act, const float* __restrict__ W,
                            const float* __restrict__ bias, float* __restrict__ out,
                            int ntok, int M, int K, int do_relu, int residual) {
  extern __shared__ float lw[];            // 64 x (K+4)
  int nTiles = ntok >> 4;
  int gw = blockIdx.x << 3;                // first wave-tile of this block
  int fB = gw / nTiles;                    // same for all 8 waves
  int nT0 = gw % nTiles;
  int featBase = fB << 6;
  int stride = K + 4;

  // async cooperative load of the 64 x K weight slab (16B chunks per lane)
  {
    int chunksPerRow = K >> 2;             // float4 chunks per row
    int totalChunks = chunksPerRow << 6;   // 64 rows
    for (int ci = threadIdx.x; ci < totalChunks; ci += 256) {
      int row = ci / chunksPerRow;
      int c4 = (ci % chunksPerRow) << 2;
      unsigned dst = (unsigned)(uintptr_t)(lw + row * stride + c4);
      const float* src = W + (size_t)(featBase + row) * K + c4;
      asm volatile("global_load_async_to_lds_b128 %0, %1, off"
                   :: "v"(dst), "v"(src) : "memory");
    }
    asm volatile("s_wait_asynccnt 0x0" ::: "memory");
  }
  __syncthreads();

  int wave = threadIdx.x >> 5, lane = threadIdx.x & 31;
  int tokBase = (nT0 + wave) << 4;
  int r = lane & 15;
  int kh = (lane >> 4) << 1;               // 0 or 2
  const float* aPtr = act + (size_t)(tokBase + r) * K + kh;
  const float* bPtr = lw + r * stride + kh;
  int bStride = 16 * stride;

  v8f acc0 = {}, acc1 = {}, acc2 = {}, acc3 = {};
  v2f a0 = *(const v2f*)(aPtr);
  int k = 0;
  for (; k < K - 4; k += 4) {
    v2f an = *(const v2f*)(aPtr + k + 4);  // unconditional prefetch
    v2f b0 = *(const v2f*)(bPtr + k);
    v2f b1 = *(const v2f*)(bPtr + bStride + k);
    v2f b2 = *(const v2f*)(bPtr + 2 * bStride + k);
    v2f b3 = *(const v2f*)(bPtr + 3 * bStride + k);
    acc0 = wmma4(a0, b0, acc0);
    acc1 = wmma4(a0, b1, acc1);
    acc2 = wmma4(a0, b2, acc2);
    acc3 = wmma4(a0, b3, acc3);
    a0 = an;
  }
  { // peeled final k-step (k == K-4)
    v2f b0 = *(const v2f*)(bPtr + k);
    v2f b1 = *(const v2f*)(bPtr + bStride + k);
    v2f b2 = *(const v2f*)(bPtr + 2 * bStride + k);
    v2f b3 = *(const v2f*)(bPtr + 3 * bStride + k);
    acc0 = wmma4(a0, b0, acc0);
    acc1 = wmma4(a0, b1, acc1);
    acc2 = wmma4(a0, b2, acc2);
    acc3 = wmma4(a0, b3, acc3);
  }

  int n = lane & 15;
  int mOff = (lane >> 4) << 3;             // 0 or 8
  v8f accs[4] = {acc0, acc1, acc2, acc3};
#pragma unroll
  for (int f = 0; f < 4; ++f) {
    int col = featBase + 16 * f + n;
    float bv = bias[col];
#pragma unroll
    for (int i = 0; i < 8; ++i) {
      float v = accs[f][i] + bv;
      if (do_relu) v = fmaxf(v, 0.f);
      size_t oi = (size_t)(tokBase + mOff + i) * M + col;
      if (residual) out[oi] += v; else out[oi] = v;
    }
  }
}

// ---------------------------------------------------------------------------
// 5) Attention: one workgroup (8 waves) per (batch-elem, head).
//    Q/K (stride 36), V (stride 40), S (stride Tp+4) in dynamic LDS —
//    strides chosen so WMMA fragment reads are bank-conflict-free.
// ---------------------------------------------------------------------------
__global__ void attn_kernel(const float* __restrict__ qkv, float* __restrict__ outp,
                            int T, int Tp, float scale) {
  extern __shared__ float smem[];
  int bt = blockIdx.x;
  int head = blockIdx.y;
  float* Q  = smem;                 // Tp x QS
  float* Kl = Q + Tp * QS;          // Tp x QS
  float* V  = Kl + Tp * QS;         // Tp x VS
  float* S  = V + Tp * VS;          // Tp x (Tp+4)
  int SS = Tp + 4;
  int tid = threadIdx.x;
  const float* base = qkv + (size_t)bt * T * 384 + head * 32;
  for (int i = tid; i < Tp * 32; i += blockDim.x) {
    int rr = i >> 5, d = i & 31;
    float q = 0.f, k = 0.f, v = 0.f;
    if (rr < T) {
      const float* p = base + (size_t)rr * 384 + d;
      q = p[0]; k = p[128]; v = p[256];
    }
    Q[rr * QS + d] = q; Kl[rr * QS + d] = k; V[rr * VS + d] = v;
  }
  __syncthreads();

  int wave = tid >> 5, lane = tid & 31;
  int r = lane & 15;
  int kh = (lane >> 4) << 1;
  int tTiles = Tp >> 4;

  // scores = Q @ K^T * scale
  for (int t = wave; t < tTiles * tTiles; t += 8) {
    int mt = t / tTiles, nt = t % tTiles;
    v8f acc = {};
    const float* qa = Q + (size_t)(mt * 16 + r) * QS + kh;
    const float* kb = Kl + (size_t)(nt * 16 + r) * QS + kh;
#pragma unroll
    for (int k0 = 0; k0 < 32; k0 += 4) {
      v2f a = *(const v2f*)(qa + k0);
      v2f b = *(const v2f*)(kb + k0);
      acc = wmma4(a, b, acc);
    }
    int n = lane & 15, mo = (lane >> 4) << 3;
#pragma unroll
    for (int i = 0; i < 8; ++i)
      S[(size_t)(mt * 16 + mo + i) * SS + nt * 16 + n] = acc[i] * scale;
  }
  __syncthreads();

  // masked softmax per row (pad cols -> 0, pad rows -> all 0)
  for (int row = wave; row < Tp; row += 8) {
    float* sr = S + (size_t)row * SS;
    if (row >= T) {
      for (int c = lane; c < Tp; c += 32) sr[c] = 0.f;
    } else {
      float mx = -3.4e38f;
      for (int c = lane; c < T; c += 32) mx = fmaxf(mx, sr[c]);
#pragma unroll
      for (int o = 16; o; o >>= 1) mx = fmaxf(mx, __shfl_xor(mx, o, 32));
      float sum = 0.f;
      for (int c = lane; c < Tp; c += 32) {
        float e = (c < T) ? expf(sr[c] - mx) : 0.f;
        sr[c] = e;
        sum += e;
      }
#pragma unroll
      for (int o = 16; o; o >>= 1) sum += __shfl_xor(sum, o, 32);
      float inv = 1.f / sum;
      for (int c = lane; c < Tp; c += 32) sr[c] *= inv;
    }
  }
  __syncthreads();

  // O = S @ V   (Tp x 32)
  for (int t = wave; t < tTiles * 2; t += 8) {
    int mt = t >> 1, nt = t & 1;
    v8f acc = {};
    const float* sa = S + (size_t)(mt * 16 + r) * SS + kh;
    int n16 = nt * 16 + r;
    for (int k0 = 0; k0 < Tp; k0 += 4) {
      v2f a = *(const v2f*)(sa + k0);
      v2f b;
      b.x = V[(size_t)(k0 + kh) * VS + n16];
      b.y = V[(size_t)(k0 + kh + 1) * VS + n16];
      acc = wmma4(a, b, acc);
    }
    int n = lane & 15, mo = (lane >> 4) << 3;
#pragma unroll
    for (int i = 0; i < 8; ++i) {
      int row = mt * 16 + mo + i;
      if (row < T)
        outp[((size_t)bt * T + row) * CDIM + head * 32 + nt * 16 + n] = acc[i];
    }
  }
}

// ---------------------------------------------------------------------------
// 6) Channel-LN (per-token over C) + affine + residual, scattered to (b,c,s,k).
// ---------------------------------------------------------------------------
__global__ void cln_kernel(const float* __restrict__ x, const float* __restrict__ g,
                           const float* __restrict__ bb, const float* __restrict__ res,
                           float* __restrict__ outp, int ntok, int intra) {
  int wave = (blockIdx.x * blockDim.x + threadIdx.x) >> 5;
  int lane = threadIdx.x & 31;
  if (wave >= ntok) return;
  float4 v = ((const float4*)(x + (size_t)wave * CDIM))[lane];
  float s = v.x + v.y + v.z + v.w;
#pragma unroll
  for (int o = 16; o; o >>= 1) s += __shfl_xor(s, o, 32);
  float mean = s * (1.f / 128.f);
  float d0 = v.x - mean, d1 = v.y - mean, d2 = v.z - mean, d3 = v.w - mean;
  float q = d0 * d0 + d1 * d1 + d2 * d2 + d3 * d3;
#pragma unroll
  for (int o = 16; o; o >>= 1) q += __shfl_xor(q, o, 32);
  float rstd = rsqrtf(q * (1.f / 128.f) + 1e-5f);
  int b = wave / POSN;
  int rem = wave % POSN;
  int sdx, kdx;
  if (intra) { sdx = rem / KDIM; kdx = rem % KDIM; }
  else       { kdx = rem / SDIM; sdx = rem % SDIM; }
  size_t posOff = (size_t)sdx * KDIM + kdx;
  float d[4] = {d0, d1, d2, d3};
#pragma unroll
  for (int j = 0; j < 4; ++j) {
    int c = lane * 4 + j;
    size_t idx = ((size_t)b * CDIM + c) * POSN + posOff;
    outp[idx] = d[j] * rstd * g[c] + bb[c] + res[idx];
  }
}

// ---------------------------------------------------------------------------
static inline void launch_gemm(const float* act, const float* W, const float* bias,
                               float* out, int ntok, int M, int K, int relu,
                               int residual, hipStream_t stream) {
  int total = (ntok >> 4) * (M >> 6);          // multiple of 8
  size_t sh = (size_t)64 * (K + 4) * sizeof(float);
  gemm_kernel<<<total / 8, 256, sh, stream>>>(act, W, bias, out, ntok, M, K,
                                              relu, residual);
}

extern "C" void kernel_launch(void* const* d_in, const int* in_sizes, int n_in,
                              void* d_out, int out_size, void* d_ws, size_t ws_size,
                              hipStream_t stream) {
  (void)in_sizes; (void)n_in; (void)out_size; (void)ws_size;
  const float* x      = (const float*)d_in[0];
  const float* enc_w  = (const float*)d_in[1];
  const float* ln1_g  = (const float*)d_in[2];
  const float* ln1_b  = (const float*)d_in[3];
  const float* qkv_w  = (const float*)d_in[4];
  const float* qkv_b  = (const float*)d_in[5];
  const float* out_w  = (const float*)d_in[6];
  const float* out_b  = (const float*)d_in[7];
  const float* ln2_g  = (const float*)d_in[8];
  const float* ln2_b  = (const float*)d_in[9];
  const float* ffn_w1 = (const float*)d_in[10];
  const float* ffn_b1 = (const float*)d_in[11];
  const float* ffn_w2 = (const float*)d_in[12];
  const float* ffn_b2 = (const float*)d_in[13];
  const float* lin_w  = (const float*)d_in[14];
  const float* lin_b  = (const float*)d_in[15];
  const float* cn_g   = (const float*)d_in[16];
  const float* cn_b   = (const float*)d_in[17];

  float* ws = (float*)d_ws;
  const size_t SZS = (size_t)4 * CDIM * POSN;   // 3,637,248 floats
  float* stateA = ws;
  float* stateB = stateA + SZS;
  float* tok    = stateB + SZS;
  float* hbuf   = tok + SZS;
  float* qkvbuf = hbuf + SZS;                   // 28416*384 floats

  // Encoder + chunking -> stateA (b,c,s,k)
  encoder_kernel<<<NTOK, 128, 0, stream>>>(x, enc_w, stateA);

  const float attn_scale = 0.17677669529663687f;  // 1/sqrt(32)

  for (int nb = 0; nb < 2; ++nb) {
    for (int stack = 0; stack < 2; ++stack) {
      int intra = (stack == 0);
      int T  = intra ? 64 : 111;
      int Tp = intra ? 64 : 112;
      int Bt = intra ? 444 : 256;
      const float* stIn = intra ? stateA : stateB;
      float* stOut = intra ? stateB : ((nb == 1) ? (float*)d_out : stateA);
      int widx = stack * 2 + nb;                // index into (2, NB, ...)

      gather_pe_kernel<<<(NTOK * 128 + 255) / 256, 256, 0, stream>>>(
          stIn, tok, NTOK * 128, intra);

      for (int l = 0; l < 4; ++l) {
        int lidx = widx * 4 + l;                // index into (2, NB, NL, ...)
        // h = LN1(x)
        ln_kernel<<<(NTOK + 7) / 8, 256, 0, stream>>>(
            tok, ln1_g + (size_t)lidx * 128, ln1_b + (size_t)lidx * 128, hbuf, NTOK);
        // qkv = h @ Wqkv^T + b
        launch_gemm(hbuf, qkv_w + (size_t)lidx * 384 * 128,
                    qkv_b + (size_t)lidx * 384, qkvbuf, NTOK, 384, 128, 0, 0, stream);
        // attention -> hbuf (token-major, per-head slices)
        size_t shbytes = (size_t)(2 * Tp * QS + Tp * VS + Tp * (Tp + 4)) * sizeof(float);
        attn_kernel<<<dim3(Bt, 4), 256, shbytes, stream>>>(qkvbuf, hbuf, T, Tp,
                                                           attn_scale);
        // x += attn_out @ Wo^T + bo
        launch_gemm(hbuf, out_w + (size_t)lidx * 128 * 128,
                    out_b + (size_t)lidx * 128, tok, NTOK, 128, 128, 0, 1, stream);
        // h = LN2(x)
        ln_kernel<<<(NTOK + 7) / 8, 256, 0, stream>>>(
            tok, ln2_g + (size_t)lidx * 128, ln2_b + (size_t)lidx * 128, hbuf, NTOK);
        // mid = relu(h @ W1^T + b1)   (reuse qkvbuf)
        launch_gemm(hbuf, ffn_w1 + (size_t)lidx * 256 * 128,
                    ffn_b1 + (size_t)lidx * 256, qkvbuf, NTOK, 256, 128, 1, 0, stream);
        // x += mid @ W2^T + b2
        launch_gemm(qkvbuf, ffn_w2 + (size_t)lidx * 128 * 256,
                    ffn_b2 + (size_t)lidx * 128, tok, NTOK, 128, 256, 0, 1, stream);
      }
      // lin projection
      launch_gemm(tok, lin_w + (size_t)widx * 128 * 128, lin_b + (size_t)widx * 128,
                  hbuf, NTOK, 128, 128, 0, 0, stream);
      // cln + residual -> next state
      cln_kernel<<<(NTOK + 7) / 8, 256, 0, stream>>>(
          hbuf, cn_g + (size_t)widx * 128, cn_b + (size_t)widx * 128, stIn, stOut,
          NTOK, intra);
    }
  }
}